// Repro_7370163880744
// MI455X (gfx1250) — compile-verified
//
#include <hip/hip_runtime.h>
#include <hip/hip_bf16.h>
#include <math.h>

// Anti-aliased separable 2D resize (16,3,500,400) -> (16,3,256,256), fp32.
// Width pass: scalar 5-tap gather into LDS. Height pass: f32 WMMA GEMM
// out_tile(16x16) = Wh(16x36) x hbuf(36x16) using v_wmma_f32_16x16x4_f32.

#define HROWS    36      // max input rows needed per 16-output-row tile (<=34)
#define HSTRIDE  272     // hbuf row stride (floats); 2*272 % 64 == 32 -> no LDS bank conflict
#define WH_STRIDE 40     // whA row stride (floats)
#define KTAPS 5

typedef __attribute__((ext_vector_type(2))) float v2f;
typedef __attribute__((ext_vector_type(8))) float v8f;

// Exact replica of the reference _axis_idx_w math for a single output index i.
// float32 arithmetic, int32 truncation-toward-zero casts, same op order.
__device__ __forceinline__ void axis_taps(int i, int in_size, float scale, float invscale,
                                          int* __restrict__ idx, float* __restrict__ w) {
    float center = ((float)i + 0.5f) * scale;
    int xmin = (int)(center - scale + 0.5f);
    if (xmin < 0) xmin = 0;
    int xmax = (int)(center + scale + 0.5f);
    if (xmax > in_size) xmax = in_size;
    int xsize = xmax - xmin;
    if (xsize > KTAPS) xsize = KTAPS;
    float xminf = (float)xmin;
    float wsum = 0.0f;
#pragma unroll
    for (int j = 0; j < KTAPS; ++j) {
        float t = ((float)j + xminf - center + 0.5f) * invscale;
        float a = fabsf(t);
        if (a > 1.0f) a = 1.0f;
        float wj = 1.0f - a;
        wj = (j < xsize) ? wj : 0.0f;
        w[j] = wj;
        wsum += wj;
        int id = xmin + j;
        if (id > in_size - 1) id = in_size - 1;
        idx[j] = id;
    }
    float inv = 1.0f / wsum;
#pragma unroll
    for (int j = 0; j < KTAPS; ++j) w[j] *= inv;
}

__global__ __launch_bounds__(256) void resize_aa_kernel(const float* __restrict__ x,
                                                        float* __restrict__ out) {
    __shared__ float hbuf[HROWS][HSTRIDE];   // horizontally-resized rows (f32)
    __shared__ float whA[16][WH_STRIDE];     // dense zero-padded vertical weights

    const int tid   = threadIdx.x;           // 0..255
    const int slice = blockIdx.y;            // 0..47 == b*3 + c
    const int oh0   = blockIdx.x * 16;       // output row tile base

    // ---- width taps for this thread's column (ow = tid), kept in registers ----
    int   wi[KTAPS];
    float ww[KTAPS];
    axis_taps(tid, 400, 1.5625f, (float)(1.0 / 1.5625), wi, ww);

    // ---- block vertical range (uniform; every thread computes it) ----
    const float sh = 1.953125f;              // 500/256
    float c0 = ((float)oh0 + 0.5f) * sh;
    int ihBase = (int)(c0 - sh + 0.5f);
    if (ihBase < 0) ihBase = 0;
    float c1 = ((float)oh0 + 15.5f) * sh;
    int ihEnd = (int)(c1 + sh + 0.5f);
    if (ihEnd > 500) ihEnd = 500;
    int span = ihEnd - ihBase;               // <= 34 < HROWS

    // ---- vertical weight matrix whA[m][k] (threads 0..15, one output row each) ----
    if (tid < 16) {
#pragma unroll
        for (int k = 0; k < WH_STRIDE; ++k) whA[tid][k] = 0.0f;
        int   hi5[KTAPS];
        float hw5[KTAPS];
        axis_taps(oh0 + tid, 500, sh, (float)(1.0 / 1.953125), hi5, hw5);
#pragma unroll
        for (int j = 0; j < KTAPS; ++j) {
            int k = hi5[j] - ihBase;
            if (k >= 0 && k < HROWS) whA[tid][k] += hw5[j];
        }
    }
    __syncthreads();

    // ---- horizontal pass: hbuf[k][tid] = 5-tap resize of input row (ihBase+k) ----
    const float* xs = x + (size_t)slice * (500 * 400);
    for (int k = 0; k < HROWS; ++k) {
        float acc = 0.0f;
        if (k < span) {
            const float* row = xs + (size_t)(ihBase + k) * 400;
            acc = fmaf(row[wi[0]], ww[0], acc);
            acc = fmaf(row[wi[1]], ww[1], acc);
            acc = fmaf(row[wi[2]], ww[2], acc);
            acc = fmaf(row[wi[3]], ww[3], acc);
            acc = fmaf(row[wi[4]], ww[4], acc);
        }
        hbuf[k][tid] = acc;                  // rows >= span zero-filled (WMMA safety)
    }
    __syncthreads();

    // ---- vertical pass via f32 WMMA: out(16x16) = whA(16x36) x hbuf(36x16) ----
    // Fragment layouts (ISA 7.12.2, 32-bit, wave32):
    //   A 16x4:  lanes 0-15 -> M=lane, v0=K0,v1=K1;  lanes 16-31 -> M=lane-16, v0=K2,v1=K3
    //   B 4x16:  lanes 0-15 -> N=lane, v0=K0,v1=K1;  lanes 16-31 -> N=lane-16, v0=K2,v1=K3
    //   D 16x16: VGPR r: lanes 0-15 -> M=r, N=lane;  lanes 16-31 -> M=r+8, N=lane-16
    const int lane  = tid & 31;
    const int wave  = tid >> 5;              // 0..7
    const int nlane = lane & 15;
    const int hi    = lane >> 4;             // 0 or 1
    const int m     = nlane;                 // A-matrix row for this lane
    const int koff  = hi * 2;                // K sub-offset within a group of 4

#pragma unroll
    for (int t = 0; t < 2; ++t) {
        const int n0 = wave * 16 + t * 128;  // output column tile base
        v8f acc = {};
#pragma unroll
        for (int k0 = 0; k0 < HROWS; k0 += 4) {
            v2f a, b;
            a.x = whA[m][k0 + koff];
            a.y = whA[m][k0 + koff + 1];
            b.x = hbuf[k0 + koff][n0 + nlane];
            b.y = hbuf[k0 + koff + 1][n0 + nlane];
            acc = __builtin_amdgcn_wmma_f32_16x16x4_f32(
                /*neg_a=*/false, a, /*neg_b=*/false, b,
                /*c_mod=*/(short)0, acc, /*reuse_a=*/false, /*reuse_b=*/false);
        }
        float* op = out + ((size_t)slice * 256 + (size_t)oh0) * 256 + n0 + nlane;
#pragma unroll
        for (int r = 0; r < 8; ++r) {
            op[(size_t)(r + hi * 8) * 256] = acc[r];
        }
    }
}

extern "C" void kernel_launch(void* const* d_in, const int* in_sizes, int n_in,
                              void* d_out, int out_size, void* d_ws, size_t ws_size,
                              hipStream_t stream) {
    (void)in_sizes; (void)n_in; (void)d_ws; (void)ws_size; (void)out_size;
    const float* x = (const float*)d_in[0];
    float* out = (float*)d_out;
    dim3 grid(16, 48);   // 16 oh-tiles x 48 (b,c) slices
    resize_aa_kernel<<<grid, dim3(256), 0, stream>>>(x, out);
}